// ODEnet_11115375362894
// MI455X (gfx1250) — compile-verified
//
#include <hip/hip_runtime.h>

typedef float v2f __attribute__((ext_vector_type(2)));
typedef float v8f __attribute__((ext_vector_type(8)));

#define TN 4  // 16-column N tiles per wave

// ---------------------------------------------------------------------------
// Generic implicit-GEMM conv (NCHW) using V_WMMA_F32_16X16X4_F32.
// Used for the stem + residual-block convs (small share of total FLOPs).
// A 16x4 layout: lane<16 -> M=lane, K={k0,k1}; lane>=16 -> M=lane-16, K={k2,k3}
// B 4x16 layout: lane<16 -> N=lane, K={k0,k1}; lane>=16 -> N=lane-16, K={k2,k3}
// C/D: vgpr v, lane<16 -> M=v, N=lane; lane>=16 -> M=v+8, N=lane-16
// ---------------------------------------------------------------------------
__global__ __launch_bounds__(128)
void conv_wmma_kernel(const float* __restrict__ in, const float* __restrict__ wgt,
                      float* __restrict__ out,
                      const float* __restrict__ addsrc, const float* __restrict__ bias,
                      int Cin, int Hin, int Win, int Hout, int Wout,
                      int stride, int pad, int ks, int Ntot)
{
    const int lane   = threadIdx.x & 31;
    const int wave   = threadIdx.x >> 5;     // 0..3 -> co block
    const int half   = lane >> 4;            // 0: K{0,1}, 1: K{2,3}
    const int m      = lane & 15;
    const int cobase = wave * 16;
    const int HWo    = Hout * Wout;
    const int HWi    = Hin * Win;
    const int KK     = ks * ks;
    const int wstr   = Cin * KK;

    int nIdx[TN], imgv[TN], oyv[TN], oxv[TN];
#pragma unroll
    for (int t = 0; t < TN; ++t) {
        int n = blockIdx.x * (TN * 16) + t * 16 + m;
        nIdx[t] = n;
        int nc = (n < Ntot) ? n : (Ntot - 1);
        int img = nc / HWo;
        int p   = nc - img * HWo;
        int y   = p / Wout;
        int x   = p - y * Wout;
        imgv[t] = img; oyv[t] = y; oxv[t] = x;
    }

    v8f acc[TN];
#pragma unroll
    for (int t = 0; t < TN; ++t) acc[t] = {};

    const int co = cobase + m;

    for (int r = 0; r < KK; ++r) {
        const int ky = r / ks, kx = r - ky * ks;
        int  boff[TN];
        bool inb[TN];
#pragma unroll
        for (int t = 0; t < TN; ++t) {
            int iy = oyv[t] * stride - pad + ky;
            int ix = oxv[t] * stride - pad + kx;
            inb[t]  = (iy >= 0) & (iy < Hin) & (ix >= 0) & (ix < Win);
            boff[t] = (imgv[t] * Cin) * HWi + iy * Win + ix;
        }
        for (int ci0 = 0; ci0 < Cin; ci0 += 4) {
            const int ciA = ci0 + 2 * half;
            v2f a;
            a.x = (ciA     < Cin) ? wgt[co * wstr + (ciA    ) * KK + r] : 0.0f;
            a.y = (ciA + 1 < Cin) ? wgt[co * wstr + (ciA + 1) * KK + r] : 0.0f;
#pragma unroll
            for (int t = 0; t < TN; ++t) {
                v2f b;
                b.x = (inb[t] && (ciA     < Cin)) ? in[boff[t] + (ciA    ) * HWi] : 0.0f;
                b.y = (inb[t] && (ciA + 1 < Cin)) ? in[boff[t] + (ciA + 1) * HWi] : 0.0f;
                acc[t] = __builtin_amdgcn_wmma_f32_16x16x4_f32(
                    false, a, false, b, (short)0, acc[t], false, false);
            }
        }
    }

#pragma unroll
    for (int t = 0; t < TN; ++t) {
        if (nIdx[t] >= Ntot) continue;
        const int img = imgv[t], y = oyv[t], x = oxv[t];
#pragma unroll
        for (int v = 0; v < 8; ++v) {
            int c   = cobase + v + 8 * half;
            int idx = ((img * 64 + c) * Hout + y) * Wout + x;
            float val = acc[t][v];
            if (bias)   val += bias[c];
            if (addsrc) val += addsrc[idx];
            out[idx] = val;
        }
    }
}

// ---------------------------------------------------------------------------
// Specialized ODE conv: 7x7, C=64->64, 3x3, stride 1, pad 1, NHWC activations,
// weights pre-transposed to [tap r][co][ci]. All operand fetches are
// contiguous float2 (b64) loads; epilogue stores 8 consecutive floats.
// ---------------------------------------------------------------------------
__global__ __launch_bounds__(128)
void conv7_nhwc_wmma_kernel(const float* __restrict__ in, const float* __restrict__ wt,
                            float* __restrict__ out, const float* __restrict__ addsrc,
                            int Ntot)
{
    const int lane   = threadIdx.x & 31;
    const int wave   = threadIdx.x >> 5;
    const int half   = lane >> 4;
    const int m      = lane & 15;
    const int cobase = wave * 16;
    const int ciA    = 2 * half;

    int nIdx[TN], oy[TN], ox[TN];
    const float* ib[TN];
#pragma unroll
    for (int t = 0; t < TN; ++t) {
        int n = blockIdx.x * (TN * 16) + t * 16 + m;
        nIdx[t] = n;
        int nc  = (n < Ntot) ? n : (Ntot - 1);
        int img = nc / 49;
        int p   = nc - img * 49;
        oy[t]   = p / 7;
        ox[t]   = p - oy[t] * 7;
        ib[t]   = in + (long)img * 49 * 64 + ciA;
    }

    v8f acc[TN];
#pragma unroll
    for (int t = 0; t < TN; ++t) acc[t] = {};

    const float* wrow = wt + (cobase + m) * 64 + ciA;   // + r*4096 + ci0

    for (int r = 0; r < 9; ++r) {
        const int ky = r / 3, kx = r - ky * 3;
        const float* bp[TN];
        bool inb[TN];
#pragma unroll
        for (int t = 0; t < TN; ++t) {
            int iy = oy[t] - 1 + ky;
            int ix = ox[t] - 1 + kx;
            inb[t] = ((unsigned)iy < 7u) & ((unsigned)ix < 7u);
            bp[t]  = ib[t] + (iy * 7 + ix) * 64;
        }
        const float* ap = wrow + r * 4096;
        for (int ci0 = 0; ci0 < 64; ci0 += 4) {
            v2f a = *(const v2f*)(ap + ci0);            // contiguous b64 load
#pragma unroll
            for (int t = 0; t < TN; ++t) {
                v2f b = inb[t] ? *(const v2f*)(bp[t] + ci0) : (v2f){0.f, 0.f};
                acc[t] = __builtin_amdgcn_wmma_f32_16x16x4_f32(
                    false, a, false, b, (short)0, acc[t], false, false);
            }
        }
    }

#pragma unroll
    for (int t = 0; t < TN; ++t) {
        if (nIdx[t] >= Ntot) continue;
        long oidx = (long)nIdx[t] * 64 + cobase + 8 * half;  // 8 consecutive floats
#pragma unroll
        for (int v = 0; v < 8; ++v) {
            float val = acc[t][v];
            if (addsrc) val += addsrc[oidx + v];
            out[oidx + v] = val;
        }
    }
}

// ---------------------------------------------------------------------------
// GroupNorm + ReLU (NCHW), one wave32 per (image, group). Supports in == out.
// ---------------------------------------------------------------------------
__global__ __launch_bounds__(256)
void gn_relu_kernel(const float* in, float* out,
                    const float* __restrict__ gamma, const float* __restrict__ beta,
                    int C, int HW, int groups, int total)
{
    const int waveid = blockIdx.x * (blockDim.x >> 5) + (threadIdx.x >> 5);
    const int lane   = threadIdx.x & 31;
    if (waveid >= total) return;
    const int n   = waveid / groups;
    const int g   = waveid - n * groups;
    const int cpg = C / groups;
    const int len = cpg * HW;
    const long base = (long)(n * C + g * cpg) * HW;

    float s = 0.f, s2 = 0.f;
    for (int i = lane; i < len; i += 32) {
        float v = in[base + i];
        s += v; s2 += v * v;
    }
#pragma unroll
    for (int off = 16; off > 0; off >>= 1) {
        s  += __shfl_xor(s,  off, 32);
        s2 += __shfl_xor(s2, off, 32);
    }
    const float mean = s / len;
    const float var  = s2 / len - mean * mean;
    const float rs   = rsqrtf(var + 1e-5f);
    for (int i = lane; i < len; i += 32) {
        float v = in[base + i];
        int   c = g * cpg + i / HW;
        float r = (v - mean) * rs * gamma[c] + beta[c];
        out[base + i] = fmaxf(r, 0.f);
    }
}

// ---------------------------------------------------------------------------
// GroupNorm + ReLU (NHWC), one thread per (image, channel); cpg in {1,2}.
// For cpg==2 the channel pair (c, c^1) lives in adjacent lanes -> shfl_xor(1).
// Fully coalesced loads/stores in NHWC. Supports in == out.
// ---------------------------------------------------------------------------
__global__ __launch_bounds__(256)
void gn_relu_nhwc_kernel(const float* in, float* out,
                         const float* __restrict__ gamma, const float* __restrict__ beta,
                         int HW, int cpg, int total)
{
    int i = blockIdx.x * blockDim.x + threadIdx.x;
    if (i >= total) return;
    int n = i >> 6;
    int c = i & 63;
    const long base = (long)n * HW * 64 + c;

    float s = 0.f, s2 = 0.f;
    for (int j = 0; j < HW; ++j) {
        float v = in[base + (long)j * 64];
        s += v; s2 += v * v;
    }
    if (cpg == 2) {
        s  += __shfl_xor(s,  1, 32);
        s2 += __shfl_xor(s2, 1, 32);
    }
    const float len  = (float)(cpg * HW);
    const float mean = s / len;
    const float var  = s2 / len - mean * mean;
    const float rs   = rsqrtf(var + 1e-5f);
    const float ga = gamma[c], be = beta[c];
    for (int j = 0; j < HW; ++j) {
        float v = in[base + (long)j * 64];
        out[base + (long)j * 64] = fmaxf((v - mean) * rs * ga + be, 0.f);
    }
}

// ---------------------------------------------------------------------------
// Layout helpers (run once per launch; negligible traffic).
// ---------------------------------------------------------------------------
__global__ void nchw_to_nhwc_kernel(const float* __restrict__ in, float* __restrict__ out,
                                    int HW, int total)
{
    int i = blockIdx.x * blockDim.x + threadIdx.x;
    if (i >= total) return;
    int c    = i & 63;
    int rest = i >> 6;
    int hw   = rest % HW;
    int n    = rest / HW;
    out[i] = in[((long)n * 64 + c) * HW + hw];
}

// (co,ci,3,3) -> [r][co][ci]
__global__ void wtrans_kernel(const float* __restrict__ w, float* __restrict__ out, int total)
{
    int i = blockIdx.x * blockDim.x + threadIdx.x;
    if (i >= total) return;
    int ci   = i & 63;
    int rest = i >> 6;
    int co   = rest & 63;
    int r    = rest >> 6;
    out[i] = w[co * 576 + ci * 9 + r];
}

// ---------------------------------------------------------------------------
// RK4 bookkeeping (layout-agnostic elementwise).
// ---------------------------------------------------------------------------
__global__ void rk4_kernel(float* z, const float* __restrict__ k,
                           float* acc, float* zt,
                           float ca, float cz, int mode, int n)
{
    int i = blockIdx.x * blockDim.x + threadIdx.x;
    if (i >= n) return;
    float kv = k[i];
    if (mode == 0)      { acc[i]  = kv;      zt[i] = z[i] + cz * kv; }
    else if (mode == 1) { acc[i] += ca * kv; zt[i] = z[i] + cz * kv; }
    else                { z[i]   += cz * (acc[i] + kv); }
}

__global__ void pool_nhwc_kernel(const float* __restrict__ in, float* __restrict__ out,
                                 int HW, int total)
{
    int i = blockIdx.x * blockDim.x + threadIdx.x;
    if (i >= total) return;
    int n = i >> 6;
    int c = i & 63;
    const float* p = in + (long)n * HW * 64 + c;
    float s = 0.f;
    for (int j = 0; j < HW; ++j) s += p[(long)j * 64];
    out[i] = s / (float)HW;
}

__global__ void fc_kernel(const float* __restrict__ pool, const float* __restrict__ w,
                          const float* __restrict__ b, float* __restrict__ out, int total)
{
    int i = blockIdx.x * blockDim.x + threadIdx.x;
    if (i >= total) return;
    int n = i / 10, o = i - n * 10;
    float s = b[o];
#pragma unroll
    for (int c = 0; c < 64; ++c) s += pool[n * 64 + c] * w[o * 64 + c];
    out[i] = s;
}

// ---------------------------------------------------------------------------
extern "C" void kernel_launch(void* const* d_in, const int* in_sizes, int n_in,
                              void* d_out, int out_size, void* d_ws, size_t ws_size,
                              hipStream_t stream)
{
    const float* x       = (const float*)d_in[0];
    const float* conv1_w = (const float*)d_in[1];
    const float* conv1_b = (const float*)d_in[2];
    const float* r1_g1w  = (const float*)d_in[3];
    const float* r1_g1b  = (const float*)d_in[4];
    const float* r1_c1w  = (const float*)d_in[5];
    const float* r1_g2w  = (const float*)d_in[6];
    const float* r1_g2b  = (const float*)d_in[7];
    const float* r1_c2w  = (const float*)d_in[8];
    const float* r1_dw   = (const float*)d_in[9];
    const float* r2_g1w  = (const float*)d_in[10];
    const float* r2_g1b  = (const float*)d_in[11];
    const float* r2_c1w  = (const float*)d_in[12];
    const float* r2_g2w  = (const float*)d_in[13];
    const float* r2_g2b  = (const float*)d_in[14];
    const float* r2_c2w  = (const float*)d_in[15];
    const float* r2_dw   = (const float*)d_in[16];
    const float* ode_g1w = (const float*)d_in[17];
    const float* ode_g1b = (const float*)d_in[18];
    const float* ode_c1w = (const float*)d_in[19];
    const float* ode_g2w = (const float*)d_in[20];
    const float* ode_g2b = (const float*)d_in[21];
    const float* ode_c2w = (const float*)d_in[22];
    const float* fin_gw  = (const float*)d_in[23];
    const float* fin_gb  = (const float*)d_in[24];
    const float* fc_w    = (const float*)d_in[25];
    const float* fc_b    = (const float*)d_in[26];

    const int B = in_sizes[0] / (28 * 28);   // 512

    // ---- scratch layout (floats) ----
    float* ws = (float*)d_ws;
    const size_t A26 = (size_t)B * 64 * 676;
    const size_t C13 = (size_t)B * 64 * 169;
    const size_t C7  = (size_t)B * 64 * 49;
    float* act26  = ws;
    float* c13a   = act26 + A26;
    float* c13s   = c13a + C13;
    float* c13o   = c13s + C13;
    float* b7a    = c13o + C13;
    float* b7s    = b7a + C7;
    float* znchw  = b7s + C7;       // r2 output (NCHW)
    float* z      = znchw + C7;     // ODE state (NHWC) from here on
    float* zt     = z + C7;
    float* hbuf   = zt + C7;
    float* tbuf   = hbuf + C7;
    float* kbuf   = tbuf + C7;
    float* accb   = kbuf + C7;
    float* pooled = accb + C7;
    float* wode1  = pooled + (size_t)B * 64;   // 9*64*64
    float* wode2  = wode1 + 9 * 64 * 64;

    auto conv = [&](const float* in, const float* wg, float* out,
                    const float* add, const float* bias,
                    int Cin, int Hin, int Win, int Hout, int Wout,
                    int s, int p, int ks) {
        int Ntot   = B * Hout * Wout;
        int blocks = (Ntot + TN * 16 - 1) / (TN * 16);
        conv_wmma_kernel<<<blocks, 128, 0, stream>>>(
            in, wg, out, add, bias, Cin, Hin, Win, Hout, Wout, s, p, ks, Ntot);
    };
    auto gn = [&](const float* in, float* out, const float* gw, const float* gb,
                  int HW, int groups) {
        int tot    = B * groups;
        int blocks = (tot + 7) / 8;
        gn_relu_kernel<<<blocks, 256, 0, stream>>>(in, out, gw, gb, 64, HW, groups, tot);
    };
    auto gn_nhwc = [&](const float* in, float* out, const float* gw, const float* gb,
                       int cpg) {
        int tot = B * 64;
        gn_relu_nhwc_kernel<<<(tot + 255) / 256, 256, 0, stream>>>(
            in, out, gw, gb, 49, cpg, tot);
    };
    auto conv7 = [&](const float* in, const float* wg, float* out, const float* add) {
        int Ntot   = B * 49;
        int blocks = (Ntot + TN * 16 - 1) / (TN * 16);
        conv7_nhwc_wmma_kernel<<<blocks, 128, 0, stream>>>(in, wg, out, add, Ntot);
    };
    auto rk4 = [&](float ca, float cz, int mode) {
        int n = (int)C7;
        rk4_kernel<<<(n + 255) / 256, 256, 0, stream>>>(z, kbuf, accb, zt, ca, cz, mode, n);
    };

    // ---- stem: conv1 (1->64, 3x3, pad 0) + bias ----
    conv(x, conv1_w, act26, nullptr, conv1_b, 1, 28, 28, 26, 26, 1, 0, 3);

    // ---- residual block 1: 26x26 -> 13x13 ----
    gn(act26, act26, r1_g1w, r1_g1b, 676, 32);
    conv(act26, r1_c1w, c13a, nullptr, nullptr, 64, 26, 26, 13, 13, 2, 1, 3);
    conv(act26, r1_dw,  c13s, nullptr, nullptr, 64, 26, 26, 13, 13, 2, 0, 1);
    gn(c13a, c13a, r1_g2w, r1_g2b, 169, 32);
    conv(c13a, r1_c2w, c13o, c13s, nullptr, 64, 13, 13, 13, 13, 1, 1, 3);

    // ---- residual block 2: 13x13 -> 7x7 ----
    gn(c13o, c13o, r2_g1w, r2_g1b, 169, 32);
    conv(c13o, r2_c1w, b7a, nullptr, nullptr, 64, 13, 13, 7, 7, 2, 1, 3);
    conv(c13o, r2_dw,  b7s, nullptr, nullptr, 64, 13, 13, 7, 7, 2, 0, 1);
    gn(b7a, b7a, r2_g2w, r2_g2b, 49, 32);
    conv(b7a, r2_c2w, znchw, b7s, nullptr, 64, 7, 7, 7, 7, 1, 1, 3);

    // ---- switch to NHWC for the ODE loop ----
    {
        int tot = (int)C7;
        nchw_to_nhwc_kernel<<<(tot + 255) / 256, 256, 0, stream>>>(znchw, z, 49, tot);
        int wt = 9 * 64 * 64;
        wtrans_kernel<<<(wt + 255) / 256, 256, 0, stream>>>(ode_c1w, wode1, wt);
        wtrans_kernel<<<(wt + 255) / 256, 256, 0, stream>>>(ode_c2w, wode2, wt);
    }

    // ---- Neural-ODE: fixed-step RK4, dt = 6/12 = 0.5 ----
    const float dt = 0.5f;
    auto odefunc = [&](const float* zin, float* kout) {
        gn_nhwc(zin, hbuf, ode_g1w, ode_g1b, 1);     // per-channel norm + relu
        conv7(hbuf, wode1, tbuf, nullptr);
        gn_nhwc(tbuf, tbuf, ode_g2w, ode_g2b, 1);
        conv7(tbuf, wode2, kout, zin);               // + zin
    };
    for (int s = 0; s < 12; ++s) {
        odefunc(z,  kbuf); rk4(0.f, 0.5f * dt, 0);   // acc=k1,  zt=z+dt/2*k1
        odefunc(zt, kbuf); rk4(2.f, 0.5f * dt, 1);   // acc+=2k2, zt=z+dt/2*k2
        odefunc(zt, kbuf); rk4(2.f, dt,        1);   // acc+=2k3, zt=z+dt*k3
        odefunc(zt, kbuf); rk4(0.f, dt / 6.f,  2);   // z+=dt/6*(acc+k4)
    }

    // ---- final norm (groups=32 -> cpg=2) + global avg pool + FC ----
    gn_nhwc(z, z, fin_gw, fin_gb, 2);
    {
        int tot = B * 64;
        pool_nhwc_kernel<<<(tot + 255) / 256, 256, 0, stream>>>(z, pooled, 49, tot);
    }
    {
        int tot = B * 10;
        fc_kernel<<<(tot + 255) / 256, 256, 0, stream>>>(pooled, fc_w, fc_b, (float*)d_out, tot);
    }
}